// CRW_53970559041937
// MI455X (gfx1250) — compile-verified
//
#include <hip/hip_runtime.h>
#include <math.h>

// Problem constants (match reference)
static constexpr int kB    = 8;
static constexpr int kC    = 512;
static constexpr int kT    = 4;
static constexpr int kHW   = 64 * 64;      // 4096
static constexpr int kNSP  = 100;
static constexpr int kDOUT = 128;
static constexpr int kNSEG = kB * kT * kNSP;   // 3200
static constexpr int kCHUNK  = 32;             // channels per seg_sum block
static constexpr int kNCHUNK = kC / kCHUNK;    // 16
static constexpr int kLPAD   = 33;             // odd stride -> spread 64 LDS banks
static constexpr int kASTR   = 516;            // A-tile LDS stride (4*mod: conflict-free b64)

typedef float v2f __attribute__((ext_vector_type(2)));
typedef float v8f __attribute__((ext_vector_type(8)));

// ---------------------------------------------------------------------------
// Kernel 1: per-(b,t,channel-chunk) segment sums. Exclusive ownership of its
// slice of `sums`, so final write is a plain store (no global atomics, no
// pre-zeroing of the workspace needed).
// ---------------------------------------------------------------------------
__global__ __launch_bounds__(256) void seg_sum_kernel(
    const float* __restrict__ maps, const int* __restrict__ sp_mask,
    float* __restrict__ sums)
{
  __shared__ float lds[kNSP * kLPAD];
  const int tid = threadIdx.x;
  for (int i = tid; i < kNSP * kLPAD; i += 256) lds[i] = 0.0f;
  __syncthreads();

  const int blk   = blockIdx.x;
  const int chunk = blk & (kNCHUNK - 1);   // which 32-channel slice
  const int bt    = blk >> 4;              // 0..31  (b*T + t)
  const int b     = bt >> 2;
  const int t     = bt & 3;

  const int lane = tid & 31;
  const int wv   = tid >> 5;               // wave id 0..7 (wave32)

  const int* spb = sp_mask + (size_t)bt * kHW;

  for (int pb = 0; pb < kHW; pb += 128) {
    const int p = pb + lane * 4;
    const int4 sp4 = *reinterpret_cast<const int4*>(spb + p);
#pragma unroll
    for (int cc = 0; cc < 4; ++cc) {
      const int cl = wv + 8 * cc;          // local channel 0..31
      const int c  = chunk * kCHUNK + cl;
      const float* mp =
          maps + ((size_t)(b * kC + c) * kT + t) * kHW + p;
      if (pb + 128 < kHW) __builtin_prefetch(mp + 128, 0, 1);
      const float4 v = *reinterpret_cast<const float4*>(mp);
      atomicAdd(&lds[sp4.x * kLPAD + cl], v.x);
      atomicAdd(&lds[sp4.y * kLPAD + cl], v.y);
      atomicAdd(&lds[sp4.z * kLPAD + cl], v.z);
      atomicAdd(&lds[sp4.w * kLPAD + cl], v.w);
    }
  }
  __syncthreads();

  // Exclusive write-back: sums[(bt*100 + n)*512 + chunk*32 + cl]
  float* outp = sums + (size_t)bt * kNSP * kC + chunk * kCHUNK;
  for (int e = tid; e < kNSP * kCHUNK; e += 256) {
    const int n  = e >> 5;
    const int cl = e & 31;
    outp[(size_t)n * kC + cl] = lds[n * kLPAD + cl];
  }
}

// ---------------------------------------------------------------------------
// Kernel 2: per-(b,t) superpixel pixel counts (tiny: 512 KB of mask reads).
// ---------------------------------------------------------------------------
__global__ __launch_bounds__(256) void seg_cnt_kernel(
    const int* __restrict__ sp_mask, float* __restrict__ counts)
{
  __shared__ int cnt[kNSP];
  const int tid = threadIdx.x;
  if (tid < kNSP) cnt[tid] = 0;
  __syncthreads();
  const int bt = blockIdx.x;
  const int* spb = sp_mask + (size_t)bt * kHW;
  for (int p = tid; p < kHW; p += 256) atomicAdd(&cnt[spb[p]], 1);
  __syncthreads();
  if (tid < kNSP) counts[bt * kNSP + tid] = (float)cnt[tid];
}

// ---------------------------------------------------------------------------
// Kernel 3: mean -> FC (WMMA f32 16x16x4) -> L2-normalize -> (B,D,T,N) store.
// One block per 16-segment tile (3200/16 = 200 blocks), 8 waves; wave w owns
// output columns [16w, 16w+16).
// ---------------------------------------------------------------------------
__global__ __launch_bounds__(256) void fc_norm_kernel(
    const float* __restrict__ sums, const float* __restrict__ counts,
    const float* __restrict__ Wfc, float* __restrict__ outp)
{
  __shared__ float lds_a[16 * kASTR];   // mean tile, padded rows
  __shared__ float lds_o[16 * kDOUT];   // FC result tile
  __shared__ float lds_inv[16];

  const int tid = threadIdx.x;
  const int s0  = blockIdx.x * 16;

  // Stage mean tile: mean[m][c] = sums/count (0 where count==0)
  for (int e = tid; e < 16 * kC; e += 256) {
    const int m = e >> 9;
    const int c = e & (kC - 1);
    const float cv = counts[s0 + m];
    const float sv = sums[(size_t)(s0 + m) * kC + c];
    lds_a[m * kASTR + c] = (cv > 0.0f) ? (sv / cv) : 0.0f;
  }
  __syncthreads();

  const int lane = tid & 31;
  const int wv   = tid >> 5;
  const int half = lane >> 4;     // K sub-pair select per ISA A/B layout
  const int ml   = lane & 15;     // M (for A) / N (for B) index
  const int n0   = wv * 16;

  v8f acc = {0.f, 0.f, 0.f, 0.f, 0.f, 0.f, 0.f, 0.f};
  const float* wrow = Wfc + (size_t)(n0 + ml) * kC;   // W_fc[d][c], row d

  for (int k0 = 0; k0 < kC; k0 += 4) {
    v2f a, bm;
    const float2 av = *reinterpret_cast<const float2*>(
        &lds_a[ml * kASTR + k0 + 2 * half]);
    a.x = av.x; a.y = av.y;
    const float2 wb = *reinterpret_cast<const float2*>(wrow + k0 + 2 * half);
    bm.x = wb.x; bm.y = wb.y;
    // D = A(16x4) * B(4x16) + C   (8 args: neg_a, A, neg_b, B, c_mod, C, reuse_a, reuse_b)
    acc = __builtin_amdgcn_wmma_f32_16x16x4_f32(
        false, a, false, bm, (short)0, acc, false, false);
  }

  // C/D layout: VGPR r, lanes 0-15 -> M=r, lanes 16-31 -> M=r+8; N=lane&15
#pragma unroll
  for (int r = 0; r < 8; ++r) {
    const int m = r + 8 * half;
    lds_o[m * kDOUT + n0 + ml] = acc[r];
  }
  __syncthreads();

  // Cross-wave L2 norm per segment row
  if (tid < 16) {
    float ss = 0.0f;
    for (int d = 0; d < kDOUT; ++d) {
      const float x = lds_o[tid * kDOUT + d];
      ss += x * x;
    }
    lds_inv[tid] = 1.0f / fmaxf(sqrtf(ss), 1e-12f);
  }
  __syncthreads();

  // Store transposed: out[b][d][t][n]
  for (int e = tid; e < 16 * kDOUT; e += 256) {
    const int m = e >> 7;
    const int d = e & (kDOUT - 1);
    const int s = s0 + m;
    const int b = s / (kT * kNSP);
    const int t = (s / kNSP) % kT;
    const int n = s % kNSP;
    outp[((size_t)(b * kDOUT + d) * kT + t) * kNSP + n] =
        lds_o[m * kDOUT + d] * lds_inv[m];
  }
}

// ---------------------------------------------------------------------------
extern "C" void kernel_launch(void* const* d_in, const int* in_sizes, int n_in,
                              void* d_out, int out_size, void* d_ws, size_t ws_size,
                              hipStream_t stream) {
  (void)in_sizes; (void)n_in; (void)out_size; (void)ws_size;
  const float* maps    = (const float*)d_in[0];
  const int*   sp_mask = (const int*)d_in[1];
  const float* Wfc     = (const float*)d_in[2];
  // d_in[3] = max_sp_num scalar; fixed at compile time (100)
  float* out = (float*)d_out;

  float* sums   = (float*)d_ws;                     // 3200*512 f32 = 6.55 MB
  float* counts = sums + (size_t)kNSEG * kC;        // 3200 f32

  seg_sum_kernel<<<kB * kT * kNCHUNK, 256, 0, stream>>>(maps, sp_mask, sums);
  seg_cnt_kernel<<<kB * kT, 256, 0, stream>>>(sp_mask, counts);
  fc_norm_kernel<<<kNSEG / 16, 256, 0, stream>>>(sums, counts, Wfc, out);
}